// DepenL_44160853737973
// MI455X (gfx1250) — compile-verified
//
#include <hip/hip_runtime.h>

// Problem constants (fixed by the reference): B=4, C=64, H=W=128, K=3, pad=1.
#define BB 4
#define CCH 64
#define H 128
#define W 128
#define LL (H * W)
#define KK 9
#define ROWS_PER_BLOCK 2
#define TPB 256
#define TILE_ROWS 4          // 2 compute rows + 2 halo rows
#define TILE_USED 130        // 128 cols + 2 halo cols
#define TILE_W 132           // padded stride (bank spread)
#define TILE_WORDS (2 * TILE_ROWS * TILE_W)
#define OUT2_OFF (BB * CCH * LL * KK)   // 37,748,736 floats

__global__ __launch_bounds__(TPB) void
stencil_mul_kernel(const float* __restrict__ keym,
                   const float* __restrict__ qrym,
                   float* __restrict__ out) {
    __shared__ float tile[2][TILE_ROWS][TILE_W];   // [0]=key, [1]=query

    const int tid   = threadIdx.x;
    const int bpp   = H / ROWS_PER_BLOCK;          // 64 blocks per plane
    const int plane = blockIdx.x / bpp;            // b*C + c
    const int y0    = (blockIdx.x % bpp) * ROWS_PER_BLOCK;

    // --- 1) zero-fill LDS tile: provides the pad=1 zero halo -------------
    {
        float* t = &tile[0][0][0];
        for (int i = tid; i < TILE_WORDS; i += TPB) t[i] = 0.0f;
    }
    __syncthreads();

    // --- 2) async global -> LDS stage of in-bounds elements --------------
    // CDNA5 path: GLOBAL_LOAD_ASYNC_TO_LDS_B32 (ASYNCcnt), GVS addressing.
    const unsigned lds_base = (unsigned)(uintptr_t)(&tile[0][0][0]);
    const int plane_base = plane * LL;             // element offset of plane

    // key_map slots (loop-invariant -> uniform SGPR base for saddr form)
    for (int i = tid; i < TILE_ROWS * TILE_USED; i += TPB) {
        const int r  = i / TILE_USED;
        const int cc = i % TILE_USED;
        const int gy = y0 - 1 + r;
        const int gx = cc - 1;
        if ((unsigned)gy < (unsigned)H && (unsigned)gx < (unsigned)W) {
            unsigned voff = (unsigned)((plane_base + gy * W + gx) * 4);
            unsigned lds  = lds_base + (unsigned)((r * TILE_W + cc) * 4);
            asm volatile("global_load_async_to_lds_b32 %0, %1, %2"
                         :: "v"(lds), "v"(voff), "s"(keym)
                         : "memory");
        }
    }
    // query_map slots
    for (int i = tid; i < TILE_ROWS * TILE_USED; i += TPB) {
        const int r  = i / TILE_USED;
        const int cc = i % TILE_USED;
        const int gy = y0 - 1 + r;
        const int gx = cc - 1;
        if ((unsigned)gy < (unsigned)H && (unsigned)gx < (unsigned)W) {
            unsigned voff = (unsigned)((plane_base + gy * W + gx) * 4);
            unsigned lds  = lds_base +
                (unsigned)(((TILE_ROWS + r) * TILE_W + cc) * 4);
            asm volatile("global_load_async_to_lds_b32 %0, %1, %2"
                         :: "v"(lds), "v"(voff), "s"(qrym)
                         : "memory");
        }
    }
    // Drain this wave's async copies, then cross-wave barrier.
    asm volatile("s_wait_asynccnt 0" ::: "memory");
    __syncthreads();

    // --- 3) compute + perfectly coalesced NON-TEMPORAL stores ------------
    // Output is 302 MB streamed write-once (> 192 MB L2): use TH=NT stores
    // so the 9x-reused 33.5 MB input stays cache-resident.
    const int obase = (plane * LL + y0 * W) * KK;
    float* __restrict__ out1 = out + obase;
    float* __restrict__ out2 = out + OUT2_OFF + obase;

#pragma unroll
    for (int j = 0; j < KK; ++j) {
        const int o   = tid + j * TPB;      // 0 .. 2303, lane-contiguous
        const int pix = o / KK;             // pixel within strip (0..255)
        const int p   = o - pix * KK;       // tap index (0..8)
        const int py  = pix >> 7;           // row within strip (0/1)
        const int px  = pix & (W - 1);      // col (0..127)
        const int dy  = p / 3;              // 0..2
        const int dx  = p - dy * 3;         // 0..2

        // Neighbor element offset within a tile plane; center = +TILE_W+1.
        const int cen = (1 + py) * TILE_W + (1 + px);
        const int nbr = (py + dy) * TILE_W + (px + dx);

        const float kc = (&tile[0][0][0])[cen];     // key center
        const float qc = (&tile[1][0][0])[cen];     // query center
        const float kn = (&tile[0][0][0])[nbr];     // key neighbor
        const float qn = (&tile[1][0][0])[nbr];     // query neighbor

        __builtin_nontemporal_store(kn * qc, out1 + o);
        __builtin_nontemporal_store(kc * qn, out2 + o);
    }
}

extern "C" void kernel_launch(void* const* d_in, const int* in_sizes, int n_in,
                              void* d_out, int out_size, void* d_ws, size_t ws_size,
                              hipStream_t stream) {
    (void)in_sizes; (void)n_in; (void)out_size; (void)d_ws; (void)ws_size;
    const float* keym = (const float*)d_in[0];
    const float* qrym = (const float*)d_in[1];
    float* out = (float*)d_out;

    const int nblocks = BB * CCH * (H / ROWS_PER_BLOCK);   // 16384
    stencil_mul_kernel<<<nblocks, TPB, 0, stream>>>(keym, qrym, out);
}